// NTK_9818295239070
// MI455X (gfx1250) — compile-verified
//
#include <hip/hip_runtime.h>
#include <math.h>

// ---------------------------------------------------------------------------
// NTK of a 256->256->256->10 MLP between two 64-sample batches, fp32,
// built on V_WMMA_F32_16X16X4_F32 (full-precision CDNA5 matrix pipe).
// ---------------------------------------------------------------------------

typedef float v2f __attribute__((ext_vector_type(2)));
typedef float v8f __attribute__((ext_vector_type(8)));

#define DH 256   // hidden / input width (all contraction dims)
#define NS 64    // samples per batch
#define NO 10    // output dim
#define NR 640   // NS*NO rows of the g matrices

static __device__ __forceinline__ v8f wmma4(v2f a, v2f b, v8f c) {
  // D = A(16x4, f32) * B(4x16, f32) + C(16x16, f32)
  return __builtin_amdgcn_wmma_f32_16x16x4_f32(
      /*neg_a=*/false, a, /*neg_b=*/false, b,
      /*c_mod=*/(short)0, c, /*reuse_a=*/false, /*reuse_b=*/false);
}

// One wave computes a 16x16 tile of A @ B^T.
// A: [*,256] row-major, rows starting at row0. B: [*,256] row-major, rows
// starting at col0 (B's rows become the tile's columns). K = 256.
// Fragment layout (ISA 7.12.2): lane = (k>=2)*16 + row ; vgpr = k&1.
static __device__ __forceinline__ v8f abt_tile(const float* __restrict__ A,
                                               const float* __restrict__ B,
                                               int row0, int col0, int lane) {
  const int rsel = lane & 15;
  const int ksel = (lane >> 4) << 1;  // 0 or 2
  const float* pa = A + (size_t)(row0 + rsel) * DH + ksel;
  const float* pb = B + (size_t)(col0 + rsel) * DH + ksel;
  v8f acc = {0.f, 0.f, 0.f, 0.f, 0.f, 0.f, 0.f, 0.f};
#pragma unroll 8
  for (int k0 = 0; k0 < DH; k0 += 4) {
    v2f a = *(const v2f*)(pa + k0);  // 8-byte aligned (k0%4==0, ksel even)
    v2f b = *(const v2f*)(pb + k0);
    acc = wmma4(a, b, acc);
  }
  return acc;
}

// H = tanh(X @ W + bias); X:[64,256], W:[256,256] (KxN row-major), H:[64,256]
// 64 tiles (4 M-tiles x 16 N-tiles), one wave each, 4 waves/block -> 16 blocks.
__global__ void ntk_fw_layer(const float* __restrict__ X,
                             const float* __restrict__ W,
                             const float* __restrict__ bias,
                             float* __restrict__ H) {
  const int lane = threadIdx.x & 31;
  const int tile = blockIdx.x * 4 + (threadIdx.x >> 5);
  const int tm = tile >> 4;  // 0..3
  const int tn = tile & 15;  // 0..15
  const int rsel = lane & 15;
  const int ksel = (lane >> 4) << 1;
  const int col = tn * 16 + rsel;
  const float* pa = X + (size_t)(tm * 16 + rsel) * DH + ksel;
  const float* pb = W + (size_t)ksel * DH + col;  // W[k][col], k = k0+ksel+v
  v8f acc = {0.f, 0.f, 0.f, 0.f, 0.f, 0.f, 0.f, 0.f};
#pragma unroll 8
  for (int k0 = 0; k0 < DH; k0 += 4) {
    v2f a = *(const v2f*)(pa + k0);
    v2f b;
    b.x = pb[(size_t)k0 * DH];
    b.y = pb[(size_t)(k0 + 1) * DH];
    acc = wmma4(a, b, acc);
  }
  const int rbase = tm * 16 + ((lane >> 4) << 3);  // M = v + 8*(lane>=16)
  const float bb = bias[col];
#pragma unroll
  for (int v = 0; v < 8; ++v)
    H[(size_t)(rbase + v) * DH + col] = tanhf(acc[v] + bb);
}

// g2[n*10+a, j] = W3[j,a] * (1 - h2[n,j]^2)   (elementwise, 640*256 elems)
__global__ void ntk_make_g2(const float* __restrict__ h2,
                            const float* __restrict__ W3,
                            float* __restrict__ g2) {
  const int idx = blockIdx.x * blockDim.x + threadIdx.x;  // < 640*256
  const int j = idx & 255;
  const int r = idx >> 8;  // n*10 + a
  const int n = r / NO;
  const int a = r - n * NO;
  const float t = h2[(size_t)n * DH + j];
  g2[idx] = W3[(size_t)j * NO + a] * (1.f - t * t);
}

// g1 = (g2 @ W2^T) * (1 - h1^2)  ; g2:[640,256], W2:[256,256], g1:[640,256]
// 640 tiles (40 x 16), wave each -> 160 blocks of 128.
__global__ void ntk_make_g1(const float* __restrict__ g2,
                            const float* __restrict__ W2,
                            const float* __restrict__ h1,
                            float* __restrict__ g1) {
  const int lane = threadIdx.x & 31;
  const int tile = blockIdx.x * 4 + (threadIdx.x >> 5);
  const int tm = tile >> 4;  // 0..39
  const int tn = tile & 15;  // 0..15
  v8f acc = abt_tile(g2, W2, tm * 16, tn * 16, lane);
  const int col = tn * 16 + (lane & 15);
  const int rbase = tm * 16 + ((lane >> 4) << 3);
#pragma unroll
  for (int v = 0; v < 8; ++v) {
    const int r = rbase + v;
    const int n = r / NO;
    const float t = h1[(size_t)n * DH + col];
    g1[(size_t)r * DH + col] = acc[v] * (1.f - t * t);
  }
}

// S = A @ B^T for A,B:[64,256] -> S:[64,64].  16 tiles -> 4 blocks of 128.
__global__ void ntk_gram64(const float* __restrict__ A,
                           const float* __restrict__ B,
                           float* __restrict__ S) {
  const int lane = threadIdx.x & 31;
  const int tile = blockIdx.x * 4 + (threadIdx.x >> 5);
  const int tm = tile >> 2;
  const int tn = tile & 3;
  v8f acc = abt_tile(A, B, tm * 16, tn * 16, lane);
  const int col = tn * 16 + (lane & 15);
  const int rbase = tm * 16 + ((lane >> 4) << 3);
#pragma unroll
  for (int v = 0; v < 8; ++v) S[(size_t)(rbase + v) * NS + col] = acc[v];
}

// Final fused kernel: per 16x16 tile of the 640x640 index space
// (r = n*10+a, c = m*10+b) compute both Gram tiles G1 = g1a.g1b^T and
// G2 = g2a.g2b^T and combine:
//   K[n,m,a,b] = d_ab*(1+S2[n,m]) + (1+S1[n,m])*G2 + (1+S0[n,m])*G1
// 1600 tiles -> 400 blocks of 128.
__global__ void ntk_combine(const float* __restrict__ g1a,
                            const float* __restrict__ g1b,
                            const float* __restrict__ g2a,
                            const float* __restrict__ g2b,
                            const float* __restrict__ S0,
                            const float* __restrict__ S1,
                            const float* __restrict__ S2,
                            float* __restrict__ out) {
  const int lane = threadIdx.x & 31;
  const int tile = blockIdx.x * 4 + (threadIdx.x >> 5);
  const int tm = tile / 40;
  const int tn = tile - tm * 40;
  v8f acc1 = abt_tile(g1a, g1b, tm * 16, tn * 16, lane);
  v8f acc2 = abt_tile(g2a, g2b, tm * 16, tn * 16, lane);
  const int c = tn * 16 + (lane & 15);
  const int m = c / NO;
  const int b = c - m * NO;
  const int rbase = tm * 16 + ((lane >> 4) << 3);
#pragma unroll
  for (int v = 0; v < 8; ++v) {
    const int r = rbase + v;
    const int n = r / NO;
    const int a = r - n * NO;
    const int nm = n * NS + m;
    float val = (1.f + S1[nm]) * acc2[v] + (1.f + S0[nm]) * acc1[v];
    if (a == b) val += 1.f + S2[nm];
    out[(size_t)((nm * NO + a) * NO + b)] = val;
  }
}

extern "C" void kernel_launch(void* const* d_in, const int* in_sizes, int n_in,
                              void* d_out, int out_size, void* d_ws,
                              size_t ws_size, hipStream_t stream) {
  (void)in_sizes; (void)n_in; (void)out_size; (void)ws_size;
  const float* x1 = (const float*)d_in[0];
  const float* x2 = (const float*)d_in[1];
  const float* W1 = (const float*)d_in[2];
  const float* b1 = (const float*)d_in[3];
  const float* W2 = (const float*)d_in[4];
  const float* b2 = (const float*)d_in[5];
  const float* W3 = (const float*)d_in[6];
  // b3 contributes the constant delta_ab term (folded into ntk_combine).

  float* ws = (float*)d_ws;
  float* h1a = ws;                 // 64*256
  float* h2a = h1a + NS * DH;
  float* h1b = h2a + NS * DH;
  float* h2b = h1b + NS * DH;
  float* g2a = h2b + NS * DH;      // 640*256
  float* g2b = g2a + NR * DH;
  float* g1a = g2b + NR * DH;
  float* g1b = g1a + NR * DH;
  float* S0 = g1b + NR * DH;       // 64*64
  float* S1 = S0 + NS * NS;
  float* S2 = S1 + NS * NS;

  // Forward passes (layer1 for both batches, then layer2).
  ntk_fw_layer<<<16, 128, 0, stream>>>(x1, W1, b1, h1a);
  ntk_fw_layer<<<16, 128, 0, stream>>>(x2, W1, b1, h1b);
  ntk_fw_layer<<<16, 128, 0, stream>>>(h1a, W2, b2, h2a);
  ntk_fw_layer<<<16, 128, 0, stream>>>(h1b, W2, b2, h2b);

  // Backprop helper matrices.
  ntk_make_g2<<<(NR * DH) / 256, 256, 0, stream>>>(h2a, W3, g2a);
  ntk_make_g2<<<(NR * DH) / 256, 256, 0, stream>>>(h2b, W3, g2b);
  ntk_make_g1<<<160, 128, 0, stream>>>(g2a, W2, h1a, g1a);
  ntk_make_g1<<<160, 128, 0, stream>>>(g2b, W2, h1b, g1b);

  // Per-layer activation Gram matrices.
  ntk_gram64<<<4, 128, 0, stream>>>(x1, x2, S0);
  ntk_gram64<<<4, 128, 0, stream>>>(h1a, h1b, S1);
  ntk_gram64<<<4, 128, 0, stream>>>(h2a, h2b, S2);

  // Fused Gram + combine -> [64,64,10,10] output.
  ntk_combine<<<400, 128, 0, stream>>>(g1a, g1b, g2a, g2b, S0, S1, S2,
                                       (float*)d_out);
}